// StopAndHopPretrained_56642028700241
// MI455X (gfx1250) — compile-verified
//
#include <hip/hip_runtime.h>

#define B_ 256
#define T_ 2048
#define V_ 256
#define C_ 10

typedef __attribute__((ext_vector_type(2))) float v2f;
typedef __attribute__((ext_vector_type(8))) float v8f;

// ---------------------------------------------------------------------------
// Kernel 1: per batch row b, find idx[b] = first t with halting logit > 0
// (default T-1).  Chunked early-exit scan; WMMA f32 16x16x4 computes the
// 256-wide dot product for 16 timesteps at a time per wave.
// ---------------------------------------------------------------------------
__global__ __launch_bounds__(128) void halt_scan_kernel(
    const float* __restrict__ pe,   // [B,T,V]
    const float* __restrict__ mp,   // [B,T,C]
    const float* __restrict__ W,    // [V+C+1]
    const float* __restrict__ bh,   // [1]
    int* __restrict__ idxOut)       // [B]
{
  __shared__ float sW[V_ + C_ + 1];
  __shared__ float sBias;
  __shared__ int   sMin;

  const int b   = blockIdx.x;
  const int tid = threadIdx.x;

  for (int i = tid; i < V_ + C_ + 1; i += 128) sW[i] = W[i];
  if (tid == 0) { sBias = bh[0]; sMin = T_ - 1; }
  __syncthreads();

  const int wave = tid >> 5;        // 4 waves per block
  const int lane = tid & 31;
  const int m    = lane & 15;       // matrix row within tile
  const int kh   = lane >> 4;       // K half: 0 -> K0,K1 ; 1 -> K2,K3

  const float bias = sBias;
  const float wT   = sW[V_ + C_];

  for (int chunk = 0; chunk < T_; chunk += 64) {
    const int t0 = chunk + wave * 16;
    const float* rowPtr =
        pe + ((size_t)b * T_ + (size_t)(t0 + m)) * V_ + 2 * kh;

    // speculative prefetch of the next chunk's corresponding row
    __builtin_prefetch(rowPtr + (size_t)64 * V_);

#if __has_builtin(__builtin_amdgcn_wmma_f32_16x16x4_f32)
    v8f c = {0.f, 0.f, 0.f, 0.f, 0.f, 0.f, 0.f, 0.f};
    #pragma unroll 8
    for (int k0 = 0; k0 < V_; k0 += 4) {
      // A: 16x4 fp32 tile of pe (ISA layout: lanes 0-15 K=0,1; lanes 16-31 K=2,3)
      v2f a = *(const v2f*)(rowPtr + k0);
      // B: 4x16, W values broadcast across all 16 columns (mirrored K layout)
      v2f bb;
      bb.x = sW[k0 + 2 * kh];
      bb.y = sW[k0 + 2 * kh + 1];
      c = __builtin_amdgcn_wmma_f32_16x16x4_f32(
              false, a, false, bb, (short)0, c, false, false);
    }
    // Every column of D is identical; column 0 lives in lanes 0 (rows 0-7,
    // VGPR r -> row r) and 16 (rows 8-15).
    if (m == 0) {
      int lmin = T_;
      #pragma unroll
      for (int r = 0; r < 8; ++r) {
        const int t = t0 + (kh << 3) + r;
        float s = c[r];
        const float* mpr = mp + ((size_t)b * T_ + t) * C_;
        #pragma unroll
        for (int j = 0; j < C_; ++j) s += mpr[j] * sW[V_ + j];
        s += ((float)t * (1.0f / (float)T_)) * wT + bias;
        if (s > 0.0f && t < lmin) lmin = t;
      }
      if (lmin < T_) atomicMin(&sMin, lmin);
    }
#else
    // Scalar fallback (should not be taken on gfx1250 toolchains)
    if (m == 0) {
      int lmin = T_;
      for (int r = 0; r < 8; ++r) {
        const int t = t0 + (kh << 3) + r;
        const float* row = pe + ((size_t)b * T_ + t) * V_;
        float s = 0.f;
        for (int k = 0; k < V_; ++k) s += row[k] * sW[k];
        const float* mpr = mp + ((size_t)b * T_ + t) * C_;
        for (int j = 0; j < C_; ++j) s += mpr[j] * sW[V_ + j];
        s += ((float)t * (1.0f / (float)T_)) * wT + bias;
        if (s > 0.0f && t < lmin) lmin = t;
      }
      if (lmin < T_) atomicMin(&sMin, lmin);
    }
#endif
    __syncthreads();
    // sMin is monotone decreasing and all writes from later chunks are
    // >= chunk+64, so this uniform break is race-free after one barrier.
    if (sMin < chunk + 64) break;
  }

  if (tid == 0) idxOut[b] = sMin;
}

// ---------------------------------------------------------------------------
// Kernel 2: per batch row, fix the times tail (zeros -> tail max), gather
// preds and the selected time.
// ---------------------------------------------------------------------------
__global__ __launch_bounds__(256) void gather_kernel(
    const float* __restrict__ mp,      // [B,T,C]
    const float* __restrict__ times,   // [B,T]
    const int*   __restrict__ idxIn,   // [B]
    float* __restrict__ outPreds,      // [B,C]   -> d_out[0 .. B*C)
    float* __restrict__ outTimesNorm,  // [B]     -> d_out[B*C+1 ..]
    float* __restrict__ wsTimes)       // [B]     raw out_times for the mean
{
  __shared__ float sred[256];
  const int b   = blockIdx.x;
  const int tid = threadIdx.x;

  float mx = -__builtin_inff();
  for (int t = 1 + tid; t < T_; t += 256)
    mx = fmaxf(mx, times[(size_t)b * T_ + t]);
  sred[tid] = mx;
  __syncthreads();
  for (int s = 128; s > 0; s >>= 1) {
    if (tid < s) sred[tid] = fmaxf(sred[tid], sred[tid + s]);
    __syncthreads();
  }

  const int idx = idxIn[b];
  if (tid < C_)
    outPreds[b * C_ + tid] = mp[((size_t)b * T_ + idx) * C_ + tid];

  if (tid == 0) {
    float tv = times[(size_t)b * T_ + idx];
    if (idx != 0 && tv == 0.0f) tv = sred[0];   // tail zeros replaced by tail max
    wsTimes[b]      = tv;
    outTimesNorm[b] = tv * (1.0f / (float)T_);
  }
}

// ---------------------------------------------------------------------------
// Kernel 3: mean of out_times over B (B == 256 == blockDim).
// ---------------------------------------------------------------------------
__global__ __launch_bounds__(256) void mean_kernel(
    const float* __restrict__ wsTimes, float* __restrict__ outMean)
{
  __shared__ float sred[256];
  const int tid = threadIdx.x;
  sred[tid] = wsTimes[tid];
  __syncthreads();
  for (int s = 128; s > 0; s >>= 1) {
    if (tid < s) sred[tid] += sred[tid + s];
    __syncthreads();
  }
  if (tid == 0) outMean[0] = sred[0] * (1.0f / (float)B_);
}

// ---------------------------------------------------------------------------
extern "C" void kernel_launch(void* const* d_in, const int* in_sizes, int n_in,
                              void* d_out, int out_size, void* d_ws, size_t ws_size,
                              hipStream_t stream) {
  const float* pe    = (const float*)d_in[0];  // prefix_embeddings [B,T,V]
  const float* mp    = (const float*)d_in[1];  // model_predictions [B,T,C]
  const float* times = (const float*)d_in[2];  // times [B,T]
  const float* W     = (const float*)d_in[3];  // W_halt [V+C+1]
  const float* bh    = (const float*)d_in[4];  // b_halt [1]

  float* out = (float*)d_out;                  // [B*C | mean | B] = 2817 floats
  int*   wsIdx   = (int*)d_ws;                 // [B] ints
  float* wsTimes = (float*)((char*)d_ws + 1024); // [B] floats

  halt_scan_kernel<<<B_, 128, 0, stream>>>(pe, mp, W, bh, wsIdx);
  gather_kernel<<<B_, 256, 0, stream>>>(mp, times, wsIdx,
                                        out,                 // preds
                                        out + B_ * C_ + 1,   // out_times / T
                                        wsTimes);
  mean_kernel<<<1, 256, 0, stream>>>(wsTimes, out + B_ * C_); // mean scalar
}